// GaussianSamplingPredict_39891656245799
// MI455X (gfx1250) — compile-verified
//
#include <hip/hip_runtime.h>
#include <math.h>

// B=1024 batches, C=100 classes, S=1000 samples.
// Fused per-batch: Cholesky in LDS -> f32 WMMA GEMM (Z * L^T + u) with
// async double-buffered Z tiles -> softmax over classes -> mean over samples.

#define B_BATCH 1024
#define C_DIM   100
#define S_DIM   1000
#define EPSV    1e-6f

#define NW       7              // waves per block (one 16-class N-tile per wave)
#define NTHREADS (NW * 32)
#define ASTR     101            // L-matrix LDS row stride (floats); odd -> conflict-free B reads
#define AROWS    112            // padded rows (N-tiles cover d=0..111)
#define TSTR     116            // Z/sample tile row stride (floats); %4==0, 16 rows -> 16 banks
#define NMT      63             // ceil(1000/16) sample tiles
#define NKB      (C_DIM / 4)    // 25 K-steps of 4

typedef __attribute__((ext_vector_type(2))) float v2f;
typedef __attribute__((ext_vector_type(8))) float v8f;

__global__ __launch_bounds__(NTHREADS)
void gsp_fused_chol_wmma_softmax(const float* __restrict__ u,
                                 const float* __restrict__ cov,
                                 const float* __restrict__ z,
                                 float* __restrict__ out)
{
    __shared__ float Alds[AROWS * ASTR];                    // L (lower-tri), zero padded
    __shared__ __align__(16) float zsbuf[2][16 * TSTR];     // double-buffered Z tile / samples tile
    __shared__ float wacc[NW * C_DIM];                      // per-wave mean accumulators
    __shared__ float ubuf[AROWS];                           // u[b,:], zero padded

    const int tid  = threadIdx.x;
    const int lane = tid & 31;
    const int wid  = tid >> 5;
    const int b    = blockIdx.x;

    // ---- init LDS ----
    for (int i = tid; i < AROWS * ASTR; i += NTHREADS) Alds[i] = 0.0f;
    for (int i = tid; i < NW * C_DIM;  i += NTHREADS) wacc[i] = 0.0f;
    if (tid < AROWS) ubuf[tid] = (tid < C_DIM) ? u[(size_t)b * C_DIM + tid] : 0.0f;
    __syncthreads();

    // ---- load lower triangle of cov[b] (upper stays zero) ----
    const float* covb = cov + (size_t)b * C_DIM * C_DIM;
    for (int i = tid; i < C_DIM * C_DIM; i += NTHREADS) {
        int r = i / C_DIM, c = i - r * C_DIM;
        if (c <= r) Alds[r * ASTR + c] = covb[i];
    }
    __syncthreads();
    if (tid < C_DIM) Alds[tid * ASTR + tid] += EPSV;        // cov*1.0 + eps*I
    __syncthreads();

    // ---- right-looking Cholesky on lower triangle ----
    for (int k = 0; k < C_DIM; ++k) {
        if (tid == 0) Alds[k * ASTR + k] = sqrtf(Alds[k * ASTR + k]);
        __syncthreads();
        float inv = 1.0f / Alds[k * ASTR + k];
        for (int i = k + 1 + tid; i < C_DIM; i += NTHREADS)
            Alds[i * ASTR + k] *= inv;
        __syncthreads();
        for (int i = k + 1 + tid; i < C_DIM; i += NTHREADS) {
            float lik = Alds[i * ASTR + k];
            for (int j = k + 1; j <= i; ++j)
                Alds[i * ASTR + j] -= lik * Alds[j * ASTR + k];
        }
        __syncthreads();
    }

    // ---- per-wave invariants: WMMA B fragments (L^T) hoisted into 50 VGPRs ----
    const int n     = lane & 15;           // N within tile (also A-frag M row)
    const int kh    = (lane >> 4) << 1;    // K half: 0 (lanes 0-15) or 2 (lanes 16-31)
    const int nbase = wid * 16;            // this wave's class-tile origin
    const int mb    = (lane >> 4) << 3;    // D-frag M base: 0 or 8
    const float uinit = ubuf[nbase + n];

    v2f bfrag[NKB];
    #pragma unroll
    for (int kk = 0; kk < NKB; ++kk) {
        bfrag[kk].x = Alds[(nbase + n) * ASTR + kk * 4 + kh];
        bfrag[kk].y = Alds[(nbase + n) * ASTR + kk * 4 + kh + 1];
    }

    // ---- async Z-tile issuer: every wave issues exactly 2 async ops per tile ----
    const float4* zg4 = (const float4*)z;                   // [1000][25] float4
    auto issue_tile = [&](int mt, float* dst) {
        if (tid < 200) {                                    // 400 float4 over 200 threads
            #pragma unroll
            for (int kp = 0; kp < 2; ++kp) {
                int idx = tid + kp * 200;                   // < 400
                int r = idx / 25, q = idx - r * 25;
                int s = mt * 16 + r;
                int sc = (s < S_DIM) ? s : 0;               // clamp (no branch: keep issue count uniform)
                unsigned ldsoff = (unsigned)(size_t)&dst[r * TSTR + q * 4];
                const float4* gp = zg4 + (size_t)sc * 25 + q;
                asm volatile("global_load_async_to_lds_b128 %0, %1, off"
                             :: "v"(ldsoff), "v"(gp) : "memory");
            }
        }
    };

    issue_tile(0, zsbuf[0]);                                // prologue

    // ---- fused GEMM + softmax + mean over sample tiles ----
    for (int mt = 0; mt < NMT; ++mt) {
        float* cur = zsbuf[mt & 1];
        float* nxt = zsbuf[(mt + 1) & 1];

        if (mt + 1 < NMT) {
            issue_tile(mt + 1, nxt);
            asm volatile("s_wait_asynccnt 0x2" ::: "memory");  // tile mt resident (in-order)
        } else {
            asm volatile("s_wait_asynccnt 0x0" ::: "memory");
        }
        __syncthreads();                                    // publish tile mt to all waves

        // accumulators = u[b, nbase+n] broadcast down M
        v8f acc;
        #pragma unroll
        for (int r = 0; r < 8; ++r) acc[r] = uinit;

        // samples(16x16) += Z(16x4) * L^T(4x16), f32 WMMA, K=100 = 25*4
        #pragma unroll
        for (int kk = 0; kk < NKB; ++kk) {
            v2f a;
            a.x = cur[n * TSTR + kk * 4 + kh];
            a.y = cur[n * TSTR + kk * 4 + kh + 1];
            acc = __builtin_amdgcn_wmma_f32_16x16x4_f32(
                      /*neg_a=*/false, a, /*neg_b=*/false, bfrag[kk],
                      /*c_mod=*/(short)0, acc,
                      /*reuse_a=*/false, /*reuse_b=*/false);
        }
        __syncthreads();                                    // all waves done reading cur as Z

        // scatter D fragments into samples tile: row m, col nbase+n
        #pragma unroll
        for (int r = 0; r < 8; ++r)
            cur[(mb + r) * TSTR + nbase + n] = acc[r];
        __syncthreads();

        // softmax per sample row + accumulate mean (rows strided over waves)
        for (int r = wid; r < 16; r += NW) {
            int s = mt * 16 + r;
            if (s >= S_DIM) continue;                       // wave-uniform
            int  c0    = lane * 4;                          // lanes 0..24 cover 100 cols
            bool valid = (lane < 25);
            float4 v = make_float4(0.0f, 0.0f, 0.0f, 0.0f);
            if (valid) v = *(const float4*)&cur[r * TSTR + c0];
            float x0 = valid ? v.x : -__builtin_inff();
            float x1 = valid ? v.y : -__builtin_inff();
            float x2 = valid ? v.z : -__builtin_inff();
            float x3 = valid ? v.w : -__builtin_inff();
            float m = fmaxf(fmaxf(x0, x1), fmaxf(x2, x3));
            for (int off = 16; off; off >>= 1) m = fmaxf(m, __shfl_xor(m, off, 32));
            float e0 = valid ? __expf(x0 - m) : 0.0f;
            float e1 = valid ? __expf(x1 - m) : 0.0f;
            float e2 = valid ? __expf(x2 - m) : 0.0f;
            float e3 = valid ? __expf(x3 - m) : 0.0f;
            float ssum = e0 + e1 + e2 + e3;
            for (int off = 16; off; off >>= 1) ssum += __shfl_xor(ssum, off, 32);
            float inv = 1.0f / ssum;
            if (valid) {
                float* wa = &wacc[wid * C_DIM + c0];
                wa[0] += e0 * inv;
                wa[1] += e1 * inv;
                wa[2] += e2 * inv;
                wa[3] += e3 * inv;
            }
        }
        __syncthreads();  // softmax reads of cur done before it is refilled next+1 iter
    }

    // ---- reduce per-wave accumulators, write mean ----
    for (int d = tid; d < C_DIM; d += NTHREADS) {
        float sum = 0.0f;
        for (int w = 0; w < NW; ++w) sum += wacc[w * C_DIM + d];
        out[(size_t)b * C_DIM + d] = sum * (1.0f / (float)S_DIM);
    }
}

extern "C" void kernel_launch(void* const* d_in, const int* in_sizes, int n_in,
                              void* d_out, int out_size, void* d_ws, size_t ws_size,
                              hipStream_t stream) {
    (void)in_sizes; (void)n_in; (void)d_ws; (void)ws_size; (void)out_size;
    const float* u   = (const float*)d_in[0];   // [1024,100]
    const float* cov = (const float*)d_in[1];   // [1024,100,100]
    const float* z   = (const float*)d_in[2];   // [1000,100]
    float*       out = (float*)d_out;           // [1024,100]

    dim3 grid(B_BATCH), block(NTHREADS);
    gsp_fused_chol_wmma_softmax<<<grid, block, 0, stream>>>(u, cov, z, out);
}